// NtmMemory_55052890800718
// MI455X (gfx1250) — compile-verified
//
#include <hip/hip_runtime.h>
#include <math.h>

// Problem constants (match reference)
#define B_SZ 8192
#define N_SZ 128
#define WD   64
#define C_SZ 128
#define WI_STRIDE 208   // 13*16 >= 198
#define RI_STRIDE 80    // 5*16  >= 70
#define EPSF 1e-8f

typedef __attribute__((ext_vector_type(2))) float v2f;
typedef __attribute__((ext_vector_type(8))) float v8f;

#if defined(__has_builtin)
#if __has_builtin(__builtin_amdgcn_wmma_f32_16x16x4_f32)
#define HAVE_WMMA_F32X4 1
#endif
#endif

// ---------------------------------------------------------------------------
// K0: zero the two global accumulators (S_w, S_r)
// ---------------------------------------------------------------------------
__global__ void k0_init(float* __restrict__ sums) {
    if (threadIdx.x < 2) sums[threadIdx.x] = 0.0f;
}

// ---------------------------------------------------------------------------
// K1: projections  write_info = ctrl @ W_write^T + b_write   [B,198 pad 208]
//                  read_info  = ctrl @ W_read^T  + b_read    [B,70  pad 80]
// One block = 16 batches. One wave = one 16x16 output tile, K-loop of 32
// V_WMMA_F32_16X16X4_F32 steps.  All operand loads are branchless b64.
// ---------------------------------------------------------------------------
__global__ void __launch_bounds__(256)
k1_proj(const float* __restrict__ ctrl,
        const float* __restrict__ Ww, const float* __restrict__ bw,
        const float* __restrict__ Wr, const float* __restrict__ br,
        float* __restrict__ wi, float* __restrict__ ri)
{
    __shared__ float ctrlS[16 * C_SZ];
    const int t  = threadIdx.x;
    const int b0 = blockIdx.x * 16;

    for (int i = t; i < 16 * C_SZ; i += 256) {
        int row = i >> 7, col = i & 127;
        ctrlS[i] = ctrl[(size_t)(b0 + row) * C_SZ + col];
    }
    __syncthreads();

    const int wave  = t >> 5;
    const int lane  = t & 31;
    const int mrow  = lane & 15;          // A-row (batch) and B-col (output)
    const int khalf = (lane >> 4) * 2;    // lanes 16-31 hold K+2,K+3
    const int mbase = (lane >> 4) * 8;    // D rows: r + 8*(lane>=16)

    for (int T = wave; T < 18; T += 8) {
        const float* W; const float* bias; float* out; int n0, nrows, stride;
        if (T < 13) { W = Ww; bias = bw; out = wi; n0 = T * 16;        nrows = 198; stride = WI_STRIDE; }
        else        { W = Wr; bias = br; out = ri; n0 = (T - 13) * 16; nrows = 70;  stride = RI_STRIDE; }

        const int   ncol  = n0 + mrow;
        const int   ncl   = (ncol < nrows) ? ncol : (nrows - 1);   // clamped, valid
        const float mask  = (ncol < nrows) ? 1.0f : 0.0f;          // branchless kill
        const float* wrow = W + (size_t)ncl * C_SZ;                // even-float aligned
        const float  bv   = bias[ncl] * mask;

        v8f c;
#pragma unroll
        for (int r = 0; r < 8; ++r) c[r] = bv;   // D[m][n] starts at bias[n]

#ifdef HAVE_WMMA_F32X4
#pragma unroll 4
        for (int k0 = 0; k0 < C_SZ; k0 += 4) {
            const int kk = k0 + khalf;                   // even
            v2f a  = *(const v2f*)&ctrlS[mrow * C_SZ + kk];   // ds_load_b64
            v2f bm = *(const v2f*)&wrow[kk];                  // global_load_b64
            bm[0] *= mask; bm[1] *= mask;
            c = __builtin_amdgcn_wmma_f32_16x16x4_f32(
                    false, a, false, bm, (short)0, c, false, false);
        }
#else
        for (int k = 0; k < C_SZ; ++k) {
            float wv = wrow[k] * mask;
#pragma unroll
            for (int r = 0; r < 8; ++r)
                c[r] += ctrlS[(mbase + r) * C_SZ + k] * wv;
        }
#endif
#pragma unroll
        for (int r = 0; r < 8; ++r)
            out[(size_t)(b0 + mbase + r) * stride + n0 + mrow] = c[r];
    }
}

// ---------------------------------------------------------------------------
// K2: write-path addressing up to un-normalized sharpened weights wg_write,
// plus global atomic accumulation of S_w.  One block (128 thr) per batch.
// Lane l owns row elements {2l, 2l+1}  ->  pure b64 streaming loads.
// ---------------------------------------------------------------------------
__global__ void __launch_bounds__(128)
k2_addr_write(const float* __restrict__ mem, const float* __restrict__ prev_w,
              const float* __restrict__ wi, float* __restrict__ wg_out,
              float* __restrict__ sums)
{
    __shared__ float keyS[WD];
    __shared__ float simS[N_SZ];
    __shared__ float wLS[N_SZ];
    __shared__ float red[N_SZ];
    __shared__ float sc[8];   // 0:keyn 1:beta 2:g 3..5:s 6:gamma
    const int t = threadIdx.x;
    const int b = blockIdx.x;
    const float* wib = wi + (size_t)b * WI_STRIDE;

    if (t < WD) keyS[t] = tanhf(wib[t]);
    if (t == 0) {
        sc[1] = fmaxf(wib[192], 0.0f);                 // beta = relu
        sc[2] = 1.0f / (1.0f + expf(-wib[193]));       // g = sigmoid
        float s0 = wib[194], s1 = wib[195], s2 = wib[196];
        float m  = fmaxf(s0, fmaxf(s1, s2));
        float e0 = expf(s0 - m), e1 = expf(s1 - m), e2 = expf(s2 - m);
        float es = e0 + e1 + e2;
        sc[3] = e0 / es; sc[4] = e1 / es; sc[5] = e2 / es;
        sc[6] = 1.0f + fmaxf(wib[197], 0.0f);          // gamma
    }
    __syncthreads();

    red[t] = (t < WD) ? keyS[t] * keyS[t] : 0.0f;
    __syncthreads();
    for (int s = 64; s > 0; s >>= 1) { if (t < s) red[t] += red[t + s]; __syncthreads(); }
    if (t == 0) sc[0] = fmaxf(sqrtf(red[0]), EPSF);
    __syncthreads();
    const float keyn = sc[0];

    // cosine similarity: one wave per row; lane l loads elements 2l,2l+1 (b64)
    const int wave = t >> 5, lane = t & 31;
    const v2f kk = *(const v2f*)&keyS[2 * lane];
    for (int it = 0; it < 32; ++it) {
        int n = it * 4 + wave;
        size_t base = ((size_t)b * N_SZ + n) * WD;
        v2f m2 = *(const v2f*)&mem[base + 2 * lane];          // global_load_b64
        float d = m2[0] * kk[0] + m2[1] * kk[1];
        float q = m2[0] * m2[0] + m2[1] * m2[1];
        for (int off = 16; off; off >>= 1) { d += __shfl_xor(d, off); q += __shfl_xor(q, off); }
        if (lane == 0) simS[n] = d / (fmaxf(sqrtf(q), EPSF) * keyn);
    }
    __syncthreads();

    // softmax(beta*sim) over n, interpolate, circular conv, sharpen
    float x = sc[1] * simS[t];
    red[t] = x; __syncthreads();
    for (int s = 64; s > 0; s >>= 1) { if (t < s) red[t] = fmaxf(red[t], red[t + s]); __syncthreads(); }
    float mx = red[0]; __syncthreads();
    float e = expf(x - mx);
    red[t] = e; __syncthreads();
    for (int s = 64; s > 0; s >>= 1) { if (t < s) red[t] += red[t + s]; __syncthreads(); }
    float sum = red[0]; __syncthreads();
    float g = sc[2];
    float w = g * (e / sum) + (1.0f - g) * prev_w[(size_t)b * N_SZ + t];
    wLS[t] = w; __syncthreads();
    float wc = sc[3] * wLS[(t + 127) & 127] + sc[4] * wLS[t] + sc[5] * wLS[(t + 1) & 127];
    float wg = powf(wc, sc[6]);
    wg_out[(size_t)b * N_SZ + t] = wg;
    red[t] = wg; __syncthreads();
    for (int s = 64; s > 0; s >>= 1) { if (t < s) red[t] += red[t + s]; __syncthreads(); }
    if (t == 0) atomicAdd(&sums[0], red[0]);
}

// ---------------------------------------------------------------------------
// K3: mem update (streams mem once via b64, writes mem' to d_out and LDS),
// read-path addressing on updated mem, atomic S_r, and the un-normalized
// read reduce acc[j] = sum_n wg_r[n]*mem'[n,j] entirely out of LDS.
// ---------------------------------------------------------------------------
__global__ void __launch_bounds__(128)
k3_update_read(const float* __restrict__ mem, const float* __restrict__ prev_w,
               const float* __restrict__ wi, const float* __restrict__ ri,
               const float* __restrict__ wg_w, float* __restrict__ sums,
               float* __restrict__ memOut, float* __restrict__ accOut)
{
    __shared__ float keyS[WD], eS[WD], aS[WD];
    __shared__ float wS[N_SZ], simS[N_SZ], red[N_SZ], wgS[N_SZ];
    __shared__ float memS[N_SZ * WD];
    __shared__ float sc[8];
    const int t = threadIdx.x;
    const int b = blockIdx.x;
    const float* wib = wi + (size_t)b * WI_STRIDE;
    const float* rib = ri + (size_t)b * RI_STRIDE;
    const float Sw = sums[0];

    if (t < WD) { keyS[t] = tanhf(rib[t]); eS[t] = wib[64 + t]; aS[t] = wib[128 + t]; }
    wS[t] = wg_w[(size_t)b * N_SZ + t] / Sw;   // finalized write weights
    if (t == 0) {
        sc[1] = fmaxf(rib[64], 0.0f);
        sc[2] = 1.0f / (1.0f + expf(-rib[65]));
        float s0 = rib[66], s1 = rib[67], s2 = rib[68];
        float m  = fmaxf(s0, fmaxf(s1, s2));
        float e0 = expf(s0 - m), e1 = expf(s1 - m), e2 = expf(s2 - m);
        float es = e0 + e1 + e2;
        sc[3] = e0 / es; sc[4] = e1 / es; sc[5] = e2 / es;
        sc[6] = 1.0f + fmaxf(rib[69], 0.0f);
    }
    __syncthreads();

    red[t] = (t < WD) ? keyS[t] * keyS[t] : 0.0f;
    __syncthreads();
    for (int s = 64; s > 0; s >>= 1) { if (t < s) red[t] += red[t + s]; __syncthreads(); }
    if (t == 0) sc[0] = fmaxf(sqrtf(red[0]), EPSF);
    __syncthreads();
    const float keyn = sc[0];

    const int wave = t >> 5, lane = t & 31;
    const v2f kk = *(const v2f*)&keyS[2 * lane];
    const v2f ee = *(const v2f*)&eS[2 * lane];
    const v2f aa = *(const v2f*)&aS[2 * lane];
    for (int it = 0; it < 32; ++it) {
        int n = it * 4 + wave;
        size_t base = ((size_t)b * N_SZ + n) * WD;
        float wn = wS[n];
        v2f m2 = *(const v2f*)&mem[base + 2 * lane];          // global_load_b64
        m2[0] = m2[0] * (1.0f - wn * ee[0]) + wn * aa[0];
        m2[1] = m2[1] * (1.0f - wn * ee[1]) + wn * aa[1];
        *(v2f*)&memOut[base + 2 * lane] = m2;                  // global_store_b64
        *(v2f*)&memS[n * WD + 2 * lane] = m2;                  // ds_store_b64
        float d = m2[0] * kk[0] + m2[1] * kk[1];
        float q = m2[0] * m2[0] + m2[1] * m2[1];
        for (int off = 16; off; off >>= 1) { d += __shfl_xor(d, off); q += __shfl_xor(q, off); }
        if (lane == 0) simS[n] = d / (fmaxf(sqrtf(q), EPSF) * keyn);
    }
    __syncthreads();

    float x = sc[1] * simS[t];
    red[t] = x; __syncthreads();
    for (int s = 64; s > 0; s >>= 1) { if (t < s) red[t] = fmaxf(red[t], red[t + s]); __syncthreads(); }
    float mx = red[0]; __syncthreads();
    float e = expf(x - mx);
    red[t] = e; __syncthreads();
    for (int s = 64; s > 0; s >>= 1) { if (t < s) red[t] += red[t + s]; __syncthreads(); }
    float sum = red[0]; __syncthreads();
    float g = sc[2];
    float w = g * (e / sum) + (1.0f - g) * prev_w[(size_t)b * N_SZ + t];
    red[t] = w; __syncthreads();                 // reuse red as w buffer for conv
    float wc = sc[3] * red[(t + 127) & 127] + sc[4] * red[t] + sc[5] * red[(t + 1) & 127];
    __syncthreads();
    float wg = powf(wc, sc[6]);
    wgS[t] = wg;
    red[t] = wg; __syncthreads();
    for (int s = 64; s > 0; s >>= 1) { if (t < s) red[t] += red[t + s]; __syncthreads(); }
    if (t == 0) atomicAdd(&sums[1], red[0]);
    __syncthreads();

    // un-normalized read vector from LDS-resident mem'
    if (t < WD) {
        float a = 0.0f;
        for (int n = 0; n < N_SZ; ++n) a += wgS[n] * memS[n * WD + t];
        accOut[(size_t)b * WD + t] = a;
    }
}

// ---------------------------------------------------------------------------
// K4: read_out = acc / S_r
// ---------------------------------------------------------------------------
__global__ void __launch_bounds__(256)
k4_finalize(const float* __restrict__ acc, const float* __restrict__ sums,
            float* __restrict__ outRead, int n)
{
    int i = blockIdx.x * blockDim.x + threadIdx.x;
    if (i < n) outRead[i] = acc[i] / sums[1];
}

// ---------------------------------------------------------------------------
extern "C" void kernel_launch(void* const* d_in, const int* in_sizes, int n_in,
                              void* d_out, int out_size, void* d_ws, size_t ws_size,
                              hipStream_t stream)
{
    const float* mem    = (const float*)d_in[0];
    const float* ctrl   = (const float*)d_in[1];
    const float* pw_w   = (const float*)d_in[2];
    const float* pw_r   = (const float*)d_in[3];
    const float* W_w    = (const float*)d_in[4];
    const float* b_w    = (const float*)d_in[5];
    const float* W_r    = (const float*)d_in[6];
    const float* b_r    = (const float*)d_in[7];

    float* memOut  = (float*)d_out;                                   // [B,N,Wd]
    float* readOut = (float*)d_out + (size_t)B_SZ * N_SZ * WD;        // [B,Wd]

    // workspace layout (floats)
    float* ws   = (float*)d_ws;
    float* wi   = ws;                                   // B * 208
    float* ri   = wi + (size_t)B_SZ * WI_STRIDE;        // B * 80
    float* wg_w = ri + (size_t)B_SZ * RI_STRIDE;        // B * 128
    float* acc  = wg_w + (size_t)B_SZ * N_SZ;           // B * 64
    float* sums = acc + (size_t)B_SZ * WD;              // 2

    k0_init<<<1, 32, 0, stream>>>(sums);
    k1_proj<<<B_SZ / 16, 256, 0, stream>>>(ctrl, W_w, b_w, W_r, b_r, wi, ri);
    k2_addr_write<<<B_SZ, 128, 0, stream>>>(mem, pw_w, wi, wg_w, sums);
    k3_update_read<<<B_SZ, 128, 0, stream>>>(mem, pw_r, wi, ri, wg_w, sums,
                                             memOut, acc);
    k4_finalize<<<(B_SZ * WD + 255) / 256, 256, 0, stream>>>(acc, sums, readOut,
                                                             B_SZ * WD);
}